// DGFA_81441169866923
// MI455X (gfx1250) — compile-verified
//
#include <hip/hip_runtime.h>
#include <hip/hip_bf16.h>

#define DIM 128
#define HEADS 8
#define HEAD_DIM 16

typedef __attribute__((ext_vector_type(2))) float    v2f;
typedef __attribute__((ext_vector_type(8))) float    v8f;
typedef __attribute__((ext_vector_type(4))) unsigned v4u;
typedef __attribute__((ext_vector_type(4))) int      v4i;
typedef __attribute__((ext_vector_type(8))) int      v8i;

// ---------- helpers ----------
__device__ __forceinline__ float leakyf(float x, float slope) {
    return x >= 0.0f ? x : slope * x;
}
// monotone float <-> uint mapping for atomicMax on floats
__device__ __forceinline__ unsigned f2ord(float f) {
    unsigned u = __float_as_uint(f);
    return (u & 0x80000000u) ? ~u : (u | 0x80000000u);
}
__device__ __forceinline__ float ord2f(unsigned u) {
    return __uint_as_float((u & 0x80000000u) ? (u & 0x7fffffffu) : ~u);
}

// ---------- 1) xh = X @ W via V_WMMA_F32_16X16X4_F32 ----------
// One wave computes a 16-row x 128-col stripe.
// W is staged TRANSPOSED with an XOR bank swizzle so each B fragment half
// (W[kk][col], W[kk+1][col]) is one aligned 8-byte LDS load:
//   Wlds[col*128 + (k ^ (2*(col&63)))] = W[k*128 + col]
// The XOR term is even, so (kk, kk+1) stay contiguous and b64-aligned, and
// the 16 lanes of each half-wave hit 16 distinct even banks.
__global__ void __launch_bounds__(128)
gat_gemm_wmma(const float* __restrict__ X, const float* __restrict__ W,
              float* __restrict__ Y, int nWaves) {
    __shared__ float Wlds[DIM * DIM];  // 64 KB, transposed + swizzled
    for (int i = threadIdx.x; i < DIM * DIM; i += blockDim.x) {
        int k = i >> 7, col = i & 127;
        Wlds[col * DIM + (k ^ (2 * (col & 63)))] = W[i];
    }
    __syncthreads();

    const int wave = blockIdx.x * (blockDim.x >> 5) + (threadIdx.x >> 5);
    const int lane = threadIdx.x & 31;
    if (wave >= nWaves) return;                 // wave-uniform exit: EXEC all-ones for WMMA
    const int row0  = wave * 16;
    const int m     = lane & 15;
    const int khalf = lane >> 4;                // 0 -> K{0,1}, 1 -> K{2,3}

    int base[8], swz[8];
#pragma unroll
    for (int t = 0; t < 8; ++t) {
        int col = t * 16 + m;
        base[t] = col * DIM;
        swz[t]  = 2 * (col & 63);
    }

    v8f acc[8] = {};
    const float* xrow = X + (size_t)(row0 + m) * DIM;

    for (int k = 0; k < DIM; k += 4) {
        const int kk = k + khalf * 2;
        v2f a = *(const v2f*)(xrow + kk);       // A[m][kk], A[m][kk+1]
#pragma unroll
        for (int t = 0; t < 8; ++t) {
            v2f b = *(const v2f*)(Wlds + base[t] + (kk ^ swz[t]));  // one ds_load_b64
            acc[t] = __builtin_amdgcn_wmma_f32_16x16x4_f32(
                false, a, false, b, (short)0, acc[t], false, false);
        }
    }

    const int mbase = (lane < 16) ? 0 : 8;
#pragma unroll
    for (int t = 0; t < 8; ++t)
#pragma unroll
        for (int v = 0; v < 8; ++v)
            Y[(size_t)(row0 + mbase + v) * DIM + t * 16 + m] = acc[t][v];
}

// ---------- 2) per-(node,head) attention logit halves ----------
__global__ void gat_alphas(const float* __restrict__ xh,
                           const float* __restrict__ a_src,
                           const float* __restrict__ a_dst,
                           float* __restrict__ as_, float* __restrict__ ad_, int N) {
    int idx = blockIdx.x * blockDim.x + threadIdx.x;   // n*HEADS + h
    if (idx >= N * HEADS) return;
    int h = idx & (HEADS - 1);
    int n = idx >> 3;
    const float4* xv = (const float4*)(xh + (size_t)n * DIM + h * HEAD_DIM);
    const float4* sv = (const float4*)(a_src + h * HEAD_DIM);
    const float4* dv = (const float4*)(a_dst + h * HEAD_DIM);
    float s = 0.f, d = 0.f;
#pragma unroll
    for (int i = 0; i < 4; ++i) {
        float4 x = xv[i], a = sv[i], b = dv[i];
        s += x.x * a.x + x.y * a.y + x.z * a.z + x.w * a.w;
        d += x.x * b.x + x.y * b.y + x.z * b.z + x.w * b.w;
    }
    as_[idx] = s;
    ad_[idx] = d;
}

// edge id -> (src,dst), self loops appended after the E real edges
__device__ __forceinline__ void edge_sd(const int* ei, int e, int E, int& s, int& d) {
    if (e < E) { s = ei[e]; d = ei[E + e]; } else { s = d = e - E; }
}

// ---------- 3) segment max of leaky(logit) per (dst, head) ----------
__global__ void gat_edge_max(const int* __restrict__ ei,
                             const float* __restrict__ as_, const float* __restrict__ ad_,
                             unsigned* __restrict__ emax, int E, int N) {
    int idx = blockIdx.x * blockDim.x + threadIdx.x;   // e*HEADS + h
    if (idx >= (E + N) * HEADS) return;
    int h = idx & (HEADS - 1);
    int e = idx >> 3;
    int s, d; edge_sd(ei, e, E, s, d);
    float v = leakyf(as_[s * HEADS + h] + ad_[d * HEADS + h], 0.2f);
    atomicMax(&emax[d * HEADS + h], f2ord(v));
}

// ---------- 4) segment sum of exp(logit - max) ----------
__global__ void gat_edge_denom(const int* __restrict__ ei,
                               const float* __restrict__ as_, const float* __restrict__ ad_,
                               const unsigned* __restrict__ emax,
                               float* __restrict__ denom, int E, int N) {
    int idx = blockIdx.x * blockDim.x + threadIdx.x;
    if (idx >= (E + N) * HEADS) return;
    int h = idx & (HEADS - 1);
    int e = idx >> 3;
    int s, d; edge_sd(ei, e, E, s, d);
    float v = leakyf(as_[s * HEADS + h] + ad_[d * HEADS + h], 0.2f);
    float m = ord2f(emax[d * HEADS + h]);
    atomicAdd(&denom[d * HEADS + h], __expf(v - m));
}

// ---------- 5) weighted scatter: out[dst] += alpha * xh[src]; wave per edge ----------
__global__ void gat_edge_aggregate(const int* __restrict__ ei,
                                   const float* __restrict__ as_, const float* __restrict__ ad_,
                                   const unsigned* __restrict__ emax,
                                   const float* __restrict__ denom,
                                   const float* __restrict__ xh,
                                   float* __restrict__ out, int E, int N) {
    int gid  = blockIdx.x * blockDim.x + threadIdx.x;
    int edge = gid >> 5;
    int lane = gid & 31;
    if (edge >= E + N) return;
    int s, d; edge_sd(ei, edge, E, s, d);
    int elem0 = lane * 4;            // 4 consecutive features per lane
    int h     = elem0 >> 4;          // head owning those features
    float v = leakyf(as_[s * HEADS + h] + ad_[d * HEADS + h], 0.2f);
    float m = ord2f(emax[d * HEADS + h]);
    float alpha = __expf(v - m) / (denom[d * HEADS + h] + 1e-16f);
    float4 xv = *(const float4*)(xh + (size_t)s * DIM + elem0);
    float* o = out + (size_t)d * DIM + elem0;
    atomicAdd(o + 0, xv.x * alpha);
    atomicAdd(o + 1, xv.y * alpha);
    atomicAdd(o + 2, xv.z * alpha);
    atomicAdd(o + 3, xv.w * alpha);
}

// ---------- 6) h = leaky(accum + bias) in place ----------
__global__ void gat_finalize(float* __restrict__ acc, const float* __restrict__ bias, int N) {
    int i = blockIdx.x * blockDim.x + threadIdx.x;
    if (i >= N * DIM) return;
    acc[i] = leakyf(acc[i] + bias[i & (DIM - 1)], 0.01f);
}

// ---------- 7) scores[n] = leaky(h@wm1+bm1)@wm2 + bm2 ----------
// wm1 (32 KB) staged into LDS with the Tensor Data Mover (TDM):
// D# group0: count=1, lds_addr, global_addr, type=2 (ISA 8.3)
// D# group1: data_size=4B, tensor_dim0=tile_dim0=8192, tile_dim1=1, stride=8192 (ISA 8.4)
// Lane 0 of wave 0 issues TENSOR_LOAD_TO_LDS (TDM ignores EXEC), waits
// s_wait_tensorcnt 0, then the block barrier releases the other waves.
__global__ void __launch_bounds__(256)
pool_scores(const float* __restrict__ h,
            const float* __restrict__ wm1, const float* __restrict__ bm1,
            const float* __restrict__ wm2, const float* __restrict__ bm2,
            float* __restrict__ scores, int N) {
    __shared__ float wm1s[DIM * 64];   // 32 KB
    if (threadIdx.x == 0) {
        unsigned long long ga = (unsigned long long)(uintptr_t)wm1;
        unsigned lds_a = (unsigned)(uintptr_t)&wm1s[0];   // flat->LDS aperture truncation
        v4u g0;
        g0.x = 1u;                                        // count=1 (valid user descriptor)
        g0.y = lds_a;                                     // lds_addr
        g0.z = (unsigned)(ga & 0xffffffffu);              // global_addr[31:0]
        g0.w = (unsigned)(((ga >> 32) & 0x01ffffffu) | 0x80000000u);  // addr[56:32] | type=2
        v8i g1;
        g1[0] = (int)(2u << 16);        // data_size = 2 -> 4 bytes
        g1[1] = (int)(8192u << 16);     // tensor_dim0[15:0] = 8192 elements
        g1[2] = (int)(1u << 16);        // tensor_dim0 hi=0, tensor_dim1 = 1
        g1[3] = (int)(8192u << 16);     // tile_dim0 = 8192
        g1[4] = 1;                      // tile_dim1 = 1, tile_dim2 = 0
        g1[5] = 8192;                   // tensor_dim0_stride = 8192
        g1[6] = 0;
        g1[7] = 0;
        v4i gz4 = {0, 0, 0, 0};         // groups 2/3 unused (<=2-D tensor)
        v8i gz8 = {0, 0, 0, 0, 0, 0, 0, 0};
        // clang-23 / therock-10.0 6-arg form: (g0, g1, g2, g3, g_extra, cpol)
        __builtin_amdgcn_tensor_load_to_lds(g0, g1, gz4, gz4, gz8, 0);
        __builtin_amdgcn_s_wait_tensorcnt(0);
    }
    __syncthreads();

    int wave   = blockIdx.x * (blockDim.x >> 5) + (threadIdx.x >> 5);
    int nWaves = gridDim.x * (blockDim.x >> 5);
    int lane   = threadIdx.x & 31;
    for (int node = wave; node < N; node += nWaves) {
        const float* row = h + (size_t)node * DIM;
        float a0 = bm1[lane], a1 = bm1[lane + 32];
        for (int k = 0; k < DIM; ++k) {
            float x = row[k];
            a0 += x * wm1s[k * 64 + lane];
            a1 += x * wm1s[k * 64 + lane + 32];
        }
        float p = leakyf(a0, 0.01f) * wm2[lane] + leakyf(a1, 0.01f) * wm2[lane + 32];
#pragma unroll
        for (int off = 16; off > 0; off >>= 1) p += __shfl_down(p, off, 32);
        if (lane == 0) scores[node] = p + bm2[0];
    }
}

// ---------- 8) global max / sum-exp of scores ----------
__global__ void reduce_max_f32(const float* __restrict__ x, unsigned* __restrict__ gmax, int n) {
    float m = -3.402823466e38f;
    for (int i = blockIdx.x * blockDim.x + threadIdx.x; i < n; i += gridDim.x * blockDim.x)
        m = fmaxf(m, x[i]);
#pragma unroll
    for (int off = 16; off > 0; off >>= 1) m = fmaxf(m, __shfl_down(m, off, 32));
    if ((threadIdx.x & 31) == 0) atomicMax(gmax, f2ord(m));
}
__global__ void reduce_sumexp(const float* __restrict__ x, const unsigned* __restrict__ gmax,
                              float* __restrict__ gsum, int n) {
    float m = ord2f(*gmax);
    float s = 0.f;
    for (int i = blockIdx.x * blockDim.x + threadIdx.x; i < n; i += gridDim.x * blockDim.x)
        s += __expf(x[i] - m);
#pragma unroll
    for (int off = 16; off > 0; off >>= 1) s += __shfl_down(s, off, 32);
    if ((threadIdx.x & 31) == 0) atomicAdd(gsum, s);
}

// ---------- 9) agg[d] = sum_n softmax(scores)[n] * h[n][d]; 128 threads/block ----------
__global__ void pool_weighted_agg(const float* __restrict__ h, const float* __restrict__ scores,
                                  const unsigned* __restrict__ gmax, const float* __restrict__ gsum,
                                  float* __restrict__ agg, int N) {
    float m = ord2f(*gmax);
    float s = *gsum;
    int d = threadIdx.x;  // 0..127
    float acc = 0.f;
    for (int n = blockIdx.x; n < N; n += gridDim.x) {
        float w = __expf(scores[n] - m) / s;
        acc += w * h[(size_t)n * DIM + d];
    }
    atomicAdd(&agg[d], acc);
}

// ---------- 10) final MLP + layernorm; single block of 128 ----------
__global__ void final_head(const float* __restrict__ agg,
                           const float* __restrict__ wa1, const float* __restrict__ ba1,
                           const float* __restrict__ wa2, const float* __restrict__ ba2,
                           const float* __restrict__ gamma, const float* __restrict__ beta,
                           float* __restrict__ out) {
    __shared__ float a[DIM];
    __shared__ float hid[64];
    __shared__ float z[DIM];
    __shared__ float red[2];
    int t = threadIdx.x;
    a[t] = agg[t];
    __syncthreads();
    if (t < 64) {
        float acc = ba1[t];
        for (int k = 0; k < DIM; ++k) acc += a[k] * wa1[k * 64 + t];
        hid[t] = leakyf(acc, 0.01f);
    }
    __syncthreads();
    float zz = ba2[t];
    for (int k = 0; k < 64; ++k) zz += hid[k] * wa2[k * DIM + t];
    z[t] = zz;
    __syncthreads();
    if (t == 0) {
        float mu = 0.f;
        for (int i = 0; i < DIM; ++i) mu += z[i];
        mu /= (float)DIM;
        float var = 0.f;
        for (int i = 0; i < DIM; ++i) { float dd = z[i] - mu; var += dd * dd; }
        var /= (float)DIM;
        red[0] = mu;
        red[1] = 1.0f / sqrtf(var + 1e-5f);
    }
    __syncthreads();
    out[t] = gamma[t] * (z[t] - red[0]) * red[1] + beta[t];
}

// =====================================================================
extern "C" void kernel_launch(void* const* d_in, const int* in_sizes, int n_in,
                              void* d_out, int out_size, void* d_ws, size_t ws_size,
                              hipStream_t stream) {
    (void)n_in; (void)out_size; (void)ws_size;

    const float* feat   = (const float*)d_in[0];
    const int*   ei     = (const int*)  d_in[1];
    const float* W1     = (const float*)d_in[2];
    const float* a_src1 = (const float*)d_in[3];
    const float* a_dst1 = (const float*)d_in[4];
    const float* b1     = (const float*)d_in[5];
    const float* W2     = (const float*)d_in[6];
    const float* a_src2 = (const float*)d_in[7];
    const float* a_dst2 = (const float*)d_in[8];
    const float* b2     = (const float*)d_in[9];
    const float* wm1    = (const float*)d_in[10];
    const float* bm1    = (const float*)d_in[11];
    const float* wm2    = (const float*)d_in[12];
    const float* bm2    = (const float*)d_in[13];
    const float* wa1    = (const float*)d_in[14];
    const float* ba1    = (const float*)d_in[15];
    const float* wa2    = (const float*)d_in[16];
    const float* ba2    = (const float*)d_in[17];
    const float* gamma  = (const float*)d_in[18];
    const float* beta   = (const float*)d_in[19];
    float* out = (float*)d_out;

    const int N = in_sizes[0] / DIM;     // 50000
    const int E = in_sizes[1] / 2;       // 600000
    const int Etot = E + N;              // with self-loops

    // ---- workspace layout (256B aligned slices) ----
    size_t off = 0;
    auto alloc = [&](size_t bytes) -> void* {
        void* p = (char*)d_ws + off;
        off += (bytes + 255) & ~(size_t)255;
        return p;
    };
    float*    bufA  = (float*)   alloc((size_t)N * DIM * sizeof(float)); // xh
    float*    bufB  = (float*)   alloc((size_t)N * DIM * sizeof(float)); // accum / h
    float*    as_   = (float*)   alloc((size_t)N * HEADS * sizeof(float));
    float*    ad_   = (float*)   alloc((size_t)N * HEADS * sizeof(float));
    unsigned* emax  = (unsigned*)alloc((size_t)N * HEADS * sizeof(unsigned));
    float*    denom = (float*)   alloc((size_t)N * HEADS * sizeof(float));
    float*    scores= (float*)   alloc((size_t)N * sizeof(float));
    unsigned* gmax  = (unsigned*)alloc(sizeof(unsigned));
    float*    gsum  = (float*)   alloc(sizeof(float));
    float*    agg   = (float*)   alloc(DIM * sizeof(float));

    const int nWaves     = N / 16;                       // 3125 row tiles
    const int gemmBlocks = (nWaves + 3) / 4;             // 4 waves / 128-thread block
    const int nhThreads  = N * HEADS;
    const int ehThreads  = Etot * HEADS;
    const int aggThreads = Etot * 32;                    // wave per edge
    const int nfThreads  = N * DIM;

    auto run_layer = [&](const float* x_in, const float* W, const float* asrc,
                         const float* adst, const float* bias) {
        // xh = x_in @ W   (fp32 WMMA)
        gat_gemm_wmma<<<gemmBlocks, 128, 0, stream>>>(x_in, W, bufA, nWaves);
        // per-(node,head) logit halves
        gat_alphas<<<(nhThreads + 255) / 256, 256, 0, stream>>>(bufA, asrc, adst, as_, ad_, N);
        // zero segment stats + accumulator (stream-ordered, capture-safe)
        (void)hipMemsetAsync(emax,  0, (size_t)N * HEADS * sizeof(unsigned), stream);
        (void)hipMemsetAsync(denom, 0, (size_t)N * HEADS * sizeof(float),    stream);
        (void)hipMemsetAsync(bufB,  0, (size_t)N * DIM   * sizeof(float),    stream);
        // segment softmax + aggregate
        gat_edge_max      <<<(ehThreads + 255) / 256, 256, 0, stream>>>(ei, as_, ad_, emax, E, N);
        gat_edge_denom    <<<(ehThreads + 255) / 256, 256, 0, stream>>>(ei, as_, ad_, emax, denom, E, N);
        gat_edge_aggregate<<<(aggThreads + 255) / 256, 256, 0, stream>>>(ei, as_, ad_, emax, denom,
                                                                         bufA, bufB, E, N);
        // h = leaky(accum + bias), in place in bufB
        gat_finalize<<<(nfThreads + 255) / 256, 256, 0, stream>>>(bufB, bias, N);
    };

    // layer 1: features -> bufB ; layer 2: bufB -> bufB
    run_layer(feat, W1, a_src1, a_dst1, b1);
    run_layer(bufB, W2, a_src2, a_dst2, b2);
    const float* h = bufB;

    // attention pooling over nodes
    pool_scores<<<512, 256, 0, stream>>>(h, wm1, bm1, wm2, bm2, scores, N);
    (void)hipMemsetAsync(gmax, 0, sizeof(unsigned), stream);
    (void)hipMemsetAsync(gsum, 0, sizeof(float), stream);
    (void)hipMemsetAsync(agg,  0, DIM * sizeof(float), stream);
    reduce_max_f32 <<<256, 256, 0, stream>>>(scores, gmax, N);
    reduce_sumexp  <<<256, 256, 0, stream>>>(scores, gmax, gsum, N);
    pool_weighted_agg<<<240, DIM, 0, stream>>>(h, scores, gmax, gsum, agg, N);

    // final MLP + layernorm -> out[128]
    final_head<<<1, DIM, 0, stream>>>(agg, wa1, ba1, wa2, ba2, gamma, beta, out);
}